// SpecialSparseConv_38981123179033
// MI455X (gfx1250) — compile-verified
//
#include <hip/hip_runtime.h>

typedef float v2f __attribute__((ext_vector_type(2)));
typedef float v8f __attribute__((ext_vector_type(8)));

#define CIN   32          // input channels == wave32 lanes
#define FOUT  64          // output channels
#define KK    9           // kernel slots
#define KCDIM (KK * CIN)  // 288 contraction length
#define TILE  16          // output rows per workgroup
#define NWAVE 4           // waves per workgroup (FOUT / 16)

__global__ __launch_bounds__(NWAVE * 32) void sparse_conv_fused_kernel(
    const float* __restrict__ feats,   // [N, 32]
    const float* __restrict__ imp,     // [N]
    const float* __restrict__ kern,    // [9,32,64] viewed as [288,64]
    const float* __restrict__ bias,    // [64]
    const int*   __restrict__ nidx,    // [E]
    const int*   __restrict__ nkidx,   // [E]
    const int*   __restrict__ splits,  // [N+1]
    float*       __restrict__ out,     // [N,64]
    float*       __restrict__ out_imp, // [N]
    int n_out)
{
    __shared__ __align__(16) float lds_bucket[TILE * KCDIM]; // 18432 B
    __shared__ float lds_imp[TILE];
    __shared__ int   lds_splits[TILE + 1];

    const int tid  = threadIdx.x;
    const int wave = tid >> 5;
    const int lane = tid & 31;
    const int tile = blockIdx.x;
    const int row0 = tile * TILE;

    // ---- zero the LDS bucket & load tile row-splits ----
    for (int i = tid; i < TILE * KCDIM; i += NWAVE * 32) lds_bucket[i] = 0.0f;
    if (tid < TILE) lds_imp[tid] = 0.0f;
    if (tid < TILE + 1) {
        int idx = row0 + tid;
        if (idx > n_out) idx = n_out;       // clamp for ragged last tile
        lds_splits[tid] = splits[idx];
    }
    __syncthreads();

    const int e_begin = lds_splits[0];
    const int e_end   = lds_splits[TILE];

    // ---- Phase 1: edge gather + weighted scatter into LDS bucket ----
    // lane == input channel; 4 waves round-robin over this tile's CSR edges.
    for (int e = e_begin + wave; e < e_end; e += NWAVE) {
        int   n = nidx[e];
        int   k = nkidx[e];
        float w = imp[n];
        // local row: largest r with splits[r] <= e  (matches searchsorted 'right'-1)
        int r = 0;
        while (r < TILE - 1 && e >= lds_splits[r + 1]) ++r;
        float v = feats[(size_t)n * CIN + lane] * w;
        atomicAdd(&lds_bucket[(r * KK + k) * CIN + lane], v);   // ds_add_f32
        if (lane == 0) atomicAdd(&lds_imp[r], w);
    }
    __syncthreads();

    // ---- importance output ----
    if (tid < TILE && row0 + tid < n_out) out_imp[row0 + tid] = lds_imp[tid];

    // ---- Phase 2: D(16x16) = A(16x288, LDS) x B(288x64 kernel), per-wave N-slab ----
    // A frag (16x4 f32, 2 VGPRs): lanes 0-15 -> M=lane, K=0/1 ; lanes 16-31 -> M=lane-16, K=2/3
    // B frag (4x16 f32, 2 VGPRs): VGPR0 = rows K=0 (lanes 0-15) / K=2 (lanes 16-31), VGPR1 = K=1 / K=3
    const int half = lane >> 4;        // 0: K pair {0,1}, 1: K pair {2,3}
    const int mcol = lane & 15;        // A row / B column / D column within slab

    v8f acc = {0.f, 0.f, 0.f, 0.f, 0.f, 0.f, 0.f, 0.f};
    const float* bptr = kern + wave * 16 + mcol;   // column base in [288,64] kernel

    #pragma unroll 4
    for (int ks = 0; ks < KCDIM / 4; ++ks) {
        const int kc = ks * 4 + half * 2;
        v2f a;
        a.x = lds_bucket[mcol * KCDIM + kc];
        a.y = lds_bucket[mcol * KCDIM + kc + 1];
        v2f b;
        b.x = bptr[(size_t)kc * FOUT];
        b.y = bptr[(size_t)(kc + 1) * FOUT];
        // 8 args: (neg_a, A, neg_b, B, c_mod, C, reuse_a, reuse_b)
        acc = __builtin_amdgcn_wmma_f32_16x16x4_f32(
            false, a, false, b, (short)0, acc, false, false);
    }

    // ---- bias + ReLU + store D tile ----
    // D layout: VGPR v -> row v (lanes 0-15) / row v+8 (lanes 16-31), col = lane&15
    const int gcol = wave * 16 + mcol;
    const float bv = bias[gcol];
    float* orow = out + (size_t)(row0 + half * 8) * FOUT + gcol;

    if (row0 + TILE <= n_out) {
        // common path: whole tile in range -> unguarded store clause
        #pragma unroll
        for (int v = 0; v < 8; ++v) {
            float val = acc[v] + bv;
            val = val > 0.f ? val : 0.f;
            orow[(size_t)v * FOUT] = val;
        }
    } else {
        // ragged last tile (unused when n_out % 16 == 0)
        #pragma unroll
        for (int v = 0; v < 8; ++v) {
            int grow = row0 + half * 8 + v;
            float val = acc[v] + bv;
            val = val > 0.f ? val : 0.f;
            if (grow < n_out) out[(size_t)grow * FOUT + gcol] = val;
        }
    }
}

extern "C" void kernel_launch(void* const* d_in, const int* in_sizes, int n_in,
                              void* d_out, int out_size, void* d_ws, size_t ws_size,
                              hipStream_t stream) {
    const float* feats  = (const float*)d_in[0];
    const float* imp    = (const float*)d_in[1];
    const float* kern   = (const float*)d_in[2];
    const float* bias   = (const float*)d_in[3];
    const int*   nidx   = (const int*)d_in[4];
    const int*   nkidx  = (const int*)d_in[5];
    const int*   splits = (const int*)d_in[6];

    const int n_out = in_sizes[1];                 // N (importance length)
    float* out     = (float*)d_out;                // [N,64]
    float* out_imp = out + (size_t)n_out * FOUT;   // [N], concatenated after out

    const int tiles = (n_out + TILE - 1) / TILE;
    sparse_conv_fused_kernel<<<tiles, NWAVE * 32, 0, stream>>>(
        feats, imp, kern, bias, nidx, nkidx, splits, out, out_imp, n_out);
}